// CrossPoolingDir_78804059947123
// MI455X (gfx1250) — compile-verified
//
#include <hip/hip_runtime.h>
#include <math.h>

// ---------------------------------------------------------------------------
// Problem constants (from reference setup_inputs):
//   gallery_value : [32, 8, 2048] f32
//   gallery_base  : [32, 8, 2048] f32
//   querys        : [64, 2048]    f32
//   W             : [2048, 2048]  f32   (out_features x in_features)
//   b,gamma,beta  : [2048]        f32
//   out           : [64, 32, 2048] f32
// ---------------------------------------------------------------------------
#define GB   32
#define GL   8
#define FEAT 2048
#define PB   64
#define MROWS (GB*GL)        // 256
#define BN_EPS 1e-5f

typedef float v2f  __attribute__((ext_vector_type(2)));
typedef float v8f  __attribute__((ext_vector_type(8)));
typedef unsigned int u32x4 __attribute__((ext_vector_type(4)));
typedef int  i32x4 __attribute__((ext_vector_type(4)));
typedef int  i32x8 __attribute__((ext_vector_type(8)));

// Flat LDS pointers carry the LDS byte offset in their low 32 bits
// (shared aperture occupies addr[63:32] only — ISA 10.2 aperture check).
static __device__ __forceinline__ unsigned lds_byte_off(const void* p) {
    return (unsigned)(uintptr_t)p;
}

// ---------------------------------------------------------------------------
// TDM: async 2D tile load Global -> LDS (TENSOR_LOAD_TO_LDS, TENSORcnt).
// D# per CDNA5 ISA §8.3/8.4:
//   group0: count=1 | lds_addr | global_addr | type=2
//   group1: data_size=4B, pad_enable (pad 4 DW after every 32 DW so each
//           32-float line lands at LDS stride 36 floats = conflict-free),
//           tensor_dim0/1, tile_dim0=32(k), tile_dim1=128(n), dim0_stride.
// ---------------------------------------------------------------------------
static __device__ __forceinline__ void tdm_load_tile_2d(
    unsigned lds_off, const void* gsrc,
    unsigned tensor_d0, unsigned tensor_d1,
    unsigned tile_d0, unsigned tile_d1,
    unsigned d0_stride)
{
    const unsigned long long ga = (unsigned long long)(uintptr_t)gsrc;

    u32x4 g0;
    g0[0] = 1u;                                        // count=1, user desc
    g0[1] = lds_off;                                   // LDS byte address
    g0[2] = (unsigned)(ga & 0xFFFFFFFFu);              // global_addr[31:0]
    g0[3] = (unsigned)((ga >> 32) & 0x01FFFFFFu)       // global_addr[56:32]
          | (2u << 30);                                // type=2 ("image")

    i32x8 g1;
    g1[0] = (int)((2u << 16)                           // data_size = 4 bytes
                | (1u << 20)                           // pad_enable
                | (4u << 22)                           // pad_interval: 32 DW
                | (3u << 25));                         // pad_amount: 4 DW
    g1[1] = (int)((tensor_d0 & 0xFFFFu) << 16);        // tensor_dim0[15:0]
    g1[2] = (int)(((tensor_d0 >> 16) & 0xFFFFu)        // tensor_dim0[31:16]
                | ((tensor_d1 & 0xFFFFu) << 16));      // tensor_dim1[15:0]
    g1[3] = (int)(((tensor_d1 >> 16) & 0xFFFFu)        // tensor_dim1[31:16]
                | ((tile_d0 & 0xFFFFu) << 16));        // tile_dim0
    g1[4] = (int)(tile_d1 & 0xFFFFu);                  // tile_dim1 (tile_dim2=0)
    g1[5] = (int)d0_stride;                            // tensor_dim0_stride[31:0]
    g1[6] = 0;                                         // stride hi / dim1_stride lo
    g1[7] = 0;

    i32x4 z4 = {0, 0, 0, 0};
#if defined(__clang_major__) && (__clang_major__ >= 23)
    i32x8 z8 = {0, 0, 0, 0, 0, 0, 0, 0};
    __builtin_amdgcn_tensor_load_to_lds(g0, g1, z4, z4, z8, 0);
#else
    __builtin_amdgcn_tensor_load_to_lds(g0, g1, z4, z4, 0);
#endif
}

// ---------------------------------------------------------------------------
// Kernel 1: Klin = base_flat[256,2048] @ W^T + bias, fp32 WMMA.
// One wave owns a 16x16 tile; 8 waves/block cover 16(M) x 128(N); K in
// chunks of 32: B tile via TDM (async, padded to stride 36), A tile via
// one float4 load + b128 DS store per thread.
// ---------------------------------------------------------------------------
#define KC  32            // k-chunk
#define KCP 36            // padded LDS row stride (floats): 144B, 16B-aligned
#define NT  128           // N per block (8 waves * 16)
#define MT  16            // M per block

__global__ __launch_bounds__(256) void gemm_bias_wmma_kernel(
    const float* __restrict__ base,   // [256, 2048]
    const float* __restrict__ Wm,     // [2048, 2048] row-major (n, k)
    const float* __restrict__ bias,   // [2048]
    float* __restrict__ Klin)         // [256, 2048]
{
    __shared__ float As[MT][KCP];     // 16 x 36
    __shared__ float Bs[NT][KCP];     // 128 x 36 (filled by TDM with padding)

    const int tid  = threadIdx.x;
    const int wave = tid >> 5;
    const int lane = tid & 31;
    const int m0   = blockIdx.y * MT;
    const int n0   = blockIdx.x * NT;

    const int hi  = (lane >= 16) ? 1 : 0;
    const int l16 = lane & 15;

    const unsigned bs_off = lds_byte_off(&Bs[0][0]);

    v8f acc = {};

    for (int kc = 0; kc < FEAT; kc += KC) {
        // ---- B tile [128 x 32] via Tensor Data Mover (wave 0 issues)
        if (tid == 0) {
            tdm_load_tile_2d(bs_off, &Wm[(size_t)n0 * FEAT + kc],
                             /*tensor_d0=*/FEAT, /*tensor_d1=*/FEAT,
                             /*tile_d0=*/KC, /*tile_d1=*/NT,
                             /*d0_stride=*/FEAT);
        }
        // ---- A tile [16 x 32]: 128 float4 transfers (b128 each way)
        if (tid < (MT * KC) / 4) {
            const int r  = tid >> 3;          // 0..15
            const int c4 = (tid & 7) * 4;     // 0,4,..,28
            const float4 v =
                *(const float4*)&base[(size_t)(m0 + r) * FEAT + kc + c4];
            *(float4*)&As[r][c4] = v;
        }
        if (tid == 0) {
            __builtin_amdgcn_s_wait_tensorcnt(0);   // TDM data landed in LDS
        }
        __syncthreads();

        // ---- 8 WMMA steps of depth 4 over this chunk
        // A 16x4: lanes 0-15 -> M=lane,K={k,k+1}; lanes 16-31 -> K={k+2,k+3}
        // B 4x16: lanes 0-15 -> N=lane,K={k,k+1}; lanes 16-31 -> K={k+2,k+3}
        #pragma unroll
        for (int k4 = 0; k4 < KC; k4 += 4) {
            const int ks = k4 + 2 * hi;
            v2f a, bfr;
            a.x   = As[l16][ks];
            a.y   = As[l16][ks + 1];
            bfr.x = Bs[wave * 16 + l16][ks];
            bfr.y = Bs[wave * 16 + l16][ks + 1];
            acc = __builtin_amdgcn_wmma_f32_16x16x4_f32(
                false, a, false, bfr, (short)0, acc, false, false);
        }
        __syncthreads();
    }

    // ---- store 16x16 tile, bias folded in
    // C/D: VGPR j, lanes 0-15 -> (M=j, N=lane); lanes 16-31 -> (M=j+8, N=lane-16)
    const int   n  = n0 + wave * 16 + l16;
    const float bv = bias[n];
    #pragma unroll
    for (int j = 0; j < 8; ++j) {
        const int m = m0 + j + hi * 8;
        Klin[(size_t)m * FEAT + n] = acc[j] + bv;
    }
}

// ---------------------------------------------------------------------------
// Kernel 2: BN training-mode stats over the 256-row axis, folded to
// scale/shift:  Khat = Klin*scale + shift.
// ---------------------------------------------------------------------------
__global__ __launch_bounds__(256) void bn_stats_kernel(
    const float* __restrict__ Klin,
    const float* __restrict__ gamma,
    const float* __restrict__ beta,
    float* __restrict__ scale,
    float* __restrict__ shift)
{
    const int f = blockIdx.x * 256 + threadIdx.x;
    if (f >= FEAT) return;
    float s = 0.f, s2 = 0.f;
    #pragma unroll 8
    for (int r = 0; r < MROWS; ++r) {
        const float v = Klin[(size_t)r * FEAT + f];
        s  += v;
        s2 += v * v;
    }
    const float inv_n = 1.0f / (float)MROWS;
    const float mu  = s * inv_n;
    const float var = fmaxf(s2 * inv_n - mu * mu, 0.0f);   // biased var
    const float sc  = gamma[f] * rsqrtf(var + BN_EPS);
    scale[f] = sc;
    shift[f] = beta[f] - mu * sc;
}

// ---------------------------------------------------------------------------
// Kernel 3: fused normalize + softmax(L=8) + weighted pool.
// Thread owns one (g,f) column: K̂[l], V[l] in registers; streams p=0..63
// with coalesced querys loads / output stores; v_exp_f32 softmax.
// ---------------------------------------------------------------------------
__global__ __launch_bounds__(256) void softmax_pool_kernel(
    const float* __restrict__ Klin,    // [256, 2048]
    const float* __restrict__ scale,   // [2048]
    const float* __restrict__ shift,   // [2048]
    const float* __restrict__ querys,  // [64, 2048]
    const float* __restrict__ gval,    // [32, 8, 2048]
    float* __restrict__ out)           // [64, 32, 2048]
{
    const int g = blockIdx.y;
    const int f = blockIdx.x * 256 + threadIdx.x;

    const float sc = scale[f];
    const float sh = shift[f];

    float kh[GL], vv[GL];
    #pragma unroll
    for (int l = 0; l < GL; ++l) {
        const int row = g * GL + l;
        kh[l] = Klin[(size_t)row * FEAT + f] * sc + sh;
        vv[l] = gval[(size_t)row * FEAT + f];
    }

    for (int p = 0; p < PB; ++p) {
        const float q = querys[(size_t)p * FEAT + f];
        float lg[GL];
        float mx = -INFINITY;
        #pragma unroll
        for (int l = 0; l < GL; ++l) {
            lg[l] = q * kh[l];
            mx = fmaxf(mx, lg[l]);
        }
        float s = 0.f, acc = 0.f;
        #pragma unroll
        for (int l = 0; l < GL; ++l) {
            const float e = __expf(lg[l] - mx);
            s   += e;
            acc += e * vv[l];
        }
        out[(size_t)(p * GB + g) * FEAT + f] = acc * (1.0f / s);
    }
}

// ---------------------------------------------------------------------------
// Launcher. d_in order: gallery_value, gallery_base, querys, W, b, gamma, beta
// Workspace: [Klin 256*2048 f32][scale 2048 f32][shift 2048 f32]
// ---------------------------------------------------------------------------
extern "C" void kernel_launch(void* const* d_in, const int* in_sizes, int n_in,
                              void* d_out, int out_size, void* d_ws, size_t ws_size,
                              hipStream_t stream) {
    const float* gval  = (const float*)d_in[0];
    const float* gbase = (const float*)d_in[1];
    const float* qrys  = (const float*)d_in[2];
    const float* Wm    = (const float*)d_in[3];
    const float* bias  = (const float*)d_in[4];
    const float* gamma = (const float*)d_in[5];
    const float* beta  = (const float*)d_in[6];
    float*       out   = (float*)d_out;

    float* Klin  = (float*)d_ws;
    float* scale = Klin + (size_t)MROWS * FEAT;
    float* shift = scale + FEAT;

    {
        dim3 grid(FEAT / NT, MROWS / MT);   // (16, 16)
        gemm_bias_wmma_kernel<<<grid, 256, 0, stream>>>(gbase, Wm, bias, Klin);
    }
    {
        dim3 grid(FEAT / 256);              // 8
        bn_stats_kernel<<<grid, 256, 0, stream>>>(Klin, gamma, beta, scale, shift);
    }
    {
        dim3 grid(FEAT / 256, GB);          // (8, 32)
        softmax_pool_kernel<<<grid, 256, 0, stream>>>(Klin, scale, shift, qrys, gval, out);
    }
}